// EdgeAtt_35742717837671
// MI455X (gfx1250) — compile-verified
//
#include <hip/hip_runtime.h>

typedef float v2f __attribute__((ext_vector_type(2)));
typedef float v8f __attribute__((ext_vector_type(8)));

#define L_DIM   110
#define D_DIM   200
#define OUT_WID 110
#define WP_     10
#define WF_     10

#define LP       112   // 7*16
#define XS_STRIDE 204  // 816B rows: 16B aligned, conflict-free 16-row column reads
#define AT_STRIDE 210  // 840B rows: 8B aligned, conflict-free 16-row column reads
#define WC_STRIDE 204  // W staging rows: 16B aligned, conflict-free b64 row reads
#define SS_STRIDE 113

#define NTHREADS 256

#define WMMA_F32X4(a, b, c) \
    __builtin_amdgcn_wmma_f32_16x16x4_f32(false, (a), false, (b), (short)0, (c), false, false)

// --- fused per-batch kernel -------------------------------------------------
__launch_bounds__(NTHREADS, 1)
__global__ void edgeatt_kernel(const float* __restrict__ X,    // [B,110,200]
                               const int*   __restrict__ lens, // [B]
                               const float* __restrict__ Wg,   // [200,200]
                               float*       __restrict__ out)  // [B,110,110]
{
    __shared__ float Xs[LP * XS_STRIDE];                 // X_b, zero-padded rows
    __shared__ float At[LP * AT_STRIDE];                 // att = X*W^T (rows=k, cols=d)
    __shared__ __align__(16) float Ss[LP * SS_STRIDE];   // scores band; stage-1 W staging alias
    __shared__ float rmax[LP];
    __shared__ float rinv[LP];

    const int b   = blockIdx.x;
    const int tid = threadIdx.x;
    const float* Xb = X + (size_t)b * (L_DIM * D_DIM);

    // ---- stage 0: coalesced float4 load of X_b into LDS, zero-pad rows 110..111
    for (int idx = tid; idx < LP * (D_DIM / 4); idx += NTHREADS) {
        int row = idx / (D_DIM / 4);
        int c4  = (idx % (D_DIM / 4)) * 4;
        float4 v = make_float4(0.f, 0.f, 0.f, 0.f);
        if (row < L_DIM) v = *(const float4*)(Xb + row * D_DIM + c4);
        *(float4*)&Xs[row * XS_STRIDE + c4] = v;
    }

    const int wave = tid >> 5;
    const int lane = tid & 31;
    const int half = lane >> 4;   // 0 or 1
    const int l16  = lane & 15;

    // ---- stage 1: att[k,d] = sum_e X[k,e] * W[d,e]   (M=112, N=208, K=200)
    // 7 rounds; each stages 32 rows of W (2 N-tiles, coalesced row-major loads)
    // into LDS, then waves 0..6 compute one 16-row M-tile against both N-tiles,
    // reusing the A fragment across the pair (register blocking).
    float* Wc = Ss;  // stage-1 alias of the scores buffer (32 x WC_STRIDE floats)
    for (int r = 0; r < 7; ++r) {
        const int nt0  = r * 2;
        const int d0   = nt0 * 16;
        const bool dual = (nt0 + 1) < 13;           // round 6 has a single N-tile
        __syncthreads();   // previous round's consumers done with Wc
        for (int idx = tid; idx < 32 * (D_DIM / 4); idx += NTHREADS) {
            int row = idx / (D_DIM / 4);
            int c4  = (idx % (D_DIM / 4)) * 4;
            int d   = d0 + row;
            float4 v = make_float4(0.f, 0.f, 0.f, 0.f);
            if (d < D_DIM) v = *(const float4*)(Wg + d * D_DIM + c4);
            *(float4*)&Wc[row * WC_STRIDE + c4] = v;
        }
        __syncthreads();   // Wc ready (also covers Xs readiness on r==0)
        if (wave < 7) {
            const int k0 = wave * 16;
            v8f c0 = {0.f, 0.f, 0.f, 0.f, 0.f, 0.f, 0.f, 0.f};
            v8f c1 = {0.f, 0.f, 0.f, 0.f, 0.f, 0.f, 0.f, 0.f};
            if (dual) {
                for (int e0 = 0; e0 < D_DIM; e0 += 4) {
                    v2f a  = *(const v2f*)&Xs[(k0 + l16) * XS_STRIDE + e0 + 2 * half];
                    v2f b0 = *(const v2f*)&Wc[(l16)      * WC_STRIDE + e0 + 2 * half];
                    v2f b1 = *(const v2f*)&Wc[(16 + l16) * WC_STRIDE + e0 + 2 * half];
                    c0 = WMMA_F32X4(a, b0, c0);
                    c1 = WMMA_F32X4(a, b1, c1);
                }
            } else {
                for (int e0 = 0; e0 < D_DIM; e0 += 4) {
                    v2f a  = *(const v2f*)&Xs[(k0 + l16) * XS_STRIDE + e0 + 2 * half];
                    v2f b0 = *(const v2f*)&Wc[(l16)      * WC_STRIDE + e0 + 2 * half];
                    c0 = WMMA_F32X4(a, b0, c0);
                }
            }
            #pragma unroll
            for (int v = 0; v < 8; ++v)
                At[(k0 + v + 8 * half) * AT_STRIDE + d0 + l16] = c0[v];
            if (dual) {
                #pragma unroll
                for (int v = 0; v < 8; ++v)
                    At[(k0 + v + 8 * half) * AT_STRIDE + d0 + 16 + l16] = c1[v];
            }
        }
    }
    __syncthreads();

    // ---- stage 2: banded scores[j,k] = sum_d X[j,d] * att[k,d]
    // wave w owns j-tile w and computes its 2-3 band k-tiles (jt-1..jt+1),
    // reusing the A fragment (X rows j) across all of them.
    if (wave < 7) {
        const int jt = wave;
        const int j0 = jt * 16;
        const bool has_lo = (jt > 0);
        const bool has_hi = (jt < 6);
        v8f cl = {0.f, 0.f, 0.f, 0.f, 0.f, 0.f, 0.f, 0.f};
        v8f cm = {0.f, 0.f, 0.f, 0.f, 0.f, 0.f, 0.f, 0.f};
        v8f ch = {0.f, 0.f, 0.f, 0.f, 0.f, 0.f, 0.f, 0.f};
        const int klo0 = (jt - 1) * 16;
        const int kmi0 = jt * 16;
        const int khi0 = (jt + 1) * 16;
        for (int d0 = 0; d0 < D_DIM; d0 += 4) {
            v2f a  = *(const v2f*)&Xs[(j0 + l16) * XS_STRIDE + d0 + 2 * half];
            v2f bm = *(const v2f*)&At[(kmi0 + l16) * AT_STRIDE + d0 + 2 * half];
            cm = WMMA_F32X4(a, bm, cm);
            if (has_lo) {
                v2f bl = *(const v2f*)&At[(klo0 + l16) * AT_STRIDE + d0 + 2 * half];
                cl = WMMA_F32X4(a, bl, cl);
            }
            if (has_hi) {
                v2f bh = *(const v2f*)&At[(khi0 + l16) * AT_STRIDE + d0 + 2 * half];
                ch = WMMA_F32X4(a, bh, ch);
            }
        }
        #pragma unroll
        for (int v = 0; v < 8; ++v)
            Ss[(j0 + v + 8 * half) * SS_STRIDE + kmi0 + l16] = cm[v];
        if (has_lo) {
            #pragma unroll
            for (int v = 0; v < 8; ++v)
                Ss[(j0 + v + 8 * half) * SS_STRIDE + klo0 + l16] = cl[v];
        }
        if (has_hi) {
            #pragma unroll
            for (int v = 0; v < 8; ++v)
                Ss[(j0 + v + 8 * half) * SS_STRIDE + khi0 + l16] = ch[v];
        }
    }
    __syncthreads();

    // ---- stage 3a: per-row band max & 1/sum (exact reference softmax semantics)
    const int len = lens[b];
    if (tid < L_DIM) {
        int j = tid;
        if (j < len) {
            int s = j - WP_; if (s < 0) s = 0;
            int e = j + WF_; if (e > len - 1) e = len - 1;
            float m = -3.0e38f;
            for (int k = s; k <= e; ++k) m = fmaxf(m, Ss[j * SS_STRIDE + k]);
            float sum = 0.f;
            for (int k = s; k <= e; ++k) sum += __expf(Ss[j * SS_STRIDE + k] - m);
            rmax[j] = m;
            rinv[j] = 1.0f / sum;
        }
    }
    __syncthreads();

    // ---- stage 3b: coalesced output write (zeros outside band / beyond len)
    float* outb = out + (size_t)b * (L_DIM * OUT_WID);
    for (int idx = tid; idx < L_DIM * OUT_WID; idx += NTHREADS) {
        int j = idx / OUT_WID;
        int k = idx - j * OUT_WID;
        float r = 0.f;
        if (j < len) {
            int s = j - WP_; if (s < 0) s = 0;
            int e = j + WF_; if (e > len - 1) e = len - 1;
            if (k >= s && k <= e)
                r = __expf(Ss[j * SS_STRIDE + k] - rmax[j]) * rinv[j];
        }
        outb[idx] = r;
    }
}

extern "C" void kernel_launch(void* const* d_in, const int* in_sizes, int n_in,
                              void* d_out, int out_size, void* d_ws, size_t ws_size,
                              hipStream_t stream) {
    const float* X    = (const float*)d_in[0];   // [B,110,200] fp32
    const int*   lens = (const int*)d_in[1];     // [B] int32
    const float* W    = (const float*)d_in[2];   // [200,200] fp32

    const int B = in_sizes[0] / (L_DIM * D_DIM); // 4096

    edgeatt_kernel<<<B, NTHREADS, 0, stream>>>(X, lens, W, (float*)d_out);
}